// LSPEGAT_57363583205984
// MI455X (gfx1250) — compile-verified
//
#include <hip/hip_runtime.h>
#include <math.h>

typedef float v2f __attribute__((ext_vector_type(2)));
typedef float v8f __attribute__((ext_vector_type(8)));

#define F 128  // feature / hidden width (fixed by the model)

// ---------- helpers: order-preserving float<->uint for atomicMax ----------
__device__ __forceinline__ unsigned f2ord(float f) {
  unsigned u = __float_as_uint(f);
  return (u & 0x80000000u) ? ~u : (u | 0x80000000u);
}
__device__ __forceinline__ float ord2f(unsigned u) {
  u = (u & 0x80000000u) ? (u & 0x7fffffffu) : ~u;
  return __uint_as_float(u);
}

// =========================================================================
// Dense transform: out[N,128] = h[N,128] @ W[128,128] via V_WMMA_F32_16X16X4_F32
// Block = 256 threads (8 waves).
// W is staged once into 64KB LDS through the CDNA5 async global->LDS DMA
// path (GLOBAL_LOAD_ASYNC_TO_LDS_B128, tracked with ASYNCcnt) -- no VGPR
// round trip for the staging traffic.
// Each wave computes a 16-row slab across all 128 output columns
// (8 tiles of 16x16, K-loop of 32 steps of K=4).
// =========================================================================
__global__ __launch_bounds__(256)
void gat_wmma_transform(const float* __restrict__ h, const float* __restrict__ W,
                        float* __restrict__ out, int nrows) {
  __shared__ float Ws[F * F];  // 64 KB

  const int tid = threadIdx.x;

  // ---- async-stage W (row-major 128x128 f32) into LDS, 16B per lane per op.
  // Flat shared pointers carry the LDS byte offset in addr[31:0], which is
  // exactly what the async-load VDST operand wants.
  {
    const unsigned lds0 = (unsigned)(size_t)(&Ws[0]) + (unsigned)tid * 16u;
    const unsigned long long g0 =
        (unsigned long long)(const void*)W + (unsigned long long)tid * 16ull;
    #pragma unroll
    for (int i = 0; i < (F * F * 4) / (256 * 16); ++i) {  // 16 chunks/thread
      const unsigned l = lds0 + (unsigned)(i * 256 * 16);
      const unsigned long long ga = g0 + (unsigned long long)(i * 256 * 16);
      asm volatile("global_load_async_to_lds_b128 %0, %1, off"
                   :: "v"(l), "v"(ga) : "memory");
    }
    asm volatile("s_wait_asynccnt 0x0" ::: "memory");
  }
  __syncthreads();

  const int wave = tid >> 5;
  const int lane = tid & 31;
  const int row0 = (blockIdx.x * 8 + wave) * 16;
  if (row0 >= nrows) return;  // uniform per wave -> EXEC stays all-ones for WMMA

  const int lhalf = lane & 15;
  const int hiOff = (lane >= 16) ? 2 : 0;  // K sub-offset for high half-wave
  const size_t arow = (size_t)(row0 + lhalf) * F;

  v8f c[8] = {};  // 8 accumulator tiles (16x16 f32 each)

  for (int k = 0; k < F; k += 4) {
    // A fragment 16x4 f32: lanes 0-15 hold K=k,k+1 ; lanes 16-31 hold K=k+2,k+3
    v2f a;
    a.x = h[arow + k + hiOff];
    a.y = h[arow + k + hiOff + 1];
    #pragma unroll
    for (int t = 0; t < 8; ++t) {
      const int col = t * 16 + lhalf;
      // B fragment 4x16 f32: v0 holds rows K=k / K=k+2, v1 holds rows K=k+1 / K=k+3
      v2f b;
      b.x = Ws[(k + hiOff) * F + col];
      b.y = Ws[(k + hiOff + 1) * F + col];
      c[t] = __builtin_amdgcn_wmma_f32_16x16x4_f32(
          /*neg_a=*/false, a, /*neg_b=*/false, b,
          /*c_mod=*/(short)0, c[t], /*reuse_a=*/false, /*reuse_b=*/false);
    }
  }

  // C/D layout: VGPR j -> row M=j (lanes 0-15) / M=j+8 (lanes 16-31), N = lane%16
  const int rbase = row0 + ((lane >= 16) ? 8 : 0);
  #pragma unroll
  for (int t = 0; t < 8; ++t) {
    #pragma unroll
    for (int j = 0; j < 8; ++j) {
      out[(size_t)(rbase + j) * F + t * 16 + lhalf] = c[t][j];
    }
  }
}

// =========================================================================
// Init: m = ord(-inf), denom = 0 (per node)
// =========================================================================
__global__ void gat_init_nodes(unsigned* __restrict__ m, float* __restrict__ denom, int n) {
  int i = blockIdx.x * blockDim.x + threadIdx.x;
  if (i < n) {
    m[i] = f2ord(-__builtin_huge_valf());
    denom[i] = 0.0f;
  }
}

__global__ void gat_zero_f4(float4* __restrict__ p, int n4) {
  int i = blockIdx.x * blockDim.x + threadIdx.x;
  if (i < n4) p[i] = make_float4(0.f, 0.f, 0.f, 0.f);
}

// =========================================================================
// Edge logits: one wave per edge. e = att . leaky_relu(hl[src] + hr[dst])
// then segment-max into m[dst] via ordered-uint atomicMax.
// Edges >= E are self-loops (src = dst = e - E).
// =========================================================================
__global__ __launch_bounds__(256)
void gat_edge_logits(const float* __restrict__ hl, const float* __restrict__ hr,
                     const int* __restrict__ ei, const float* __restrict__ att,
                     float* __restrict__ ebuf, unsigned* __restrict__ m,
                     int E, int Etot) {
  const int eid = blockIdx.x * 8 + (threadIdx.x >> 5);
  if (eid >= Etot) return;
  const int lane = threadIdx.x & 31;
  int s, d;
  if (eid < E) { s = ei[eid]; d = ei[E + eid]; }
  else         { s = eid - E; d = s; }

  const float4 l4 = ((const float4*)(hl + (size_t)s * F))[lane];
  const float4 r4 = ((const float4*)(hr + (size_t)d * F))[lane];
  const float4 a4 = ((const float4*)att)[lane];

  float sum = 0.f, v;
  v = l4.x + r4.x; v = (v > 0.f) ? v : 0.2f * v; sum += v * a4.x;
  v = l4.y + r4.y; v = (v > 0.f) ? v : 0.2f * v; sum += v * a4.y;
  v = l4.z + r4.z; v = (v > 0.f) ? v : 0.2f * v; sum += v * a4.z;
  v = l4.w + r4.w; v = (v > 0.f) ? v : 0.2f * v; sum += v * a4.w;

  #pragma unroll
  for (int o = 16; o > 0; o >>= 1) sum += __shfl_xor(sum, o, 32);

  if (lane == 0) {
    ebuf[eid] = sum;
    atomicMax(&m[d], f2ord(sum));
  }
}

// =========================================================================
// a = exp(e - m[dst]); denom[dst] += a   (one thread per edge)
// =========================================================================
__global__ void gat_edge_expsum(float* __restrict__ ebuf, const unsigned* __restrict__ m,
                                const int* __restrict__ ei, float* __restrict__ denom,
                                int E, int Etot) {
  const int eid = blockIdx.x * blockDim.x + threadIdx.x;
  if (eid >= Etot) return;
  const int d = (eid < E) ? ei[E + eid] : (eid - E);
  const float a = __expf(ebuf[eid] - ord2f(m[d]));
  ebuf[eid] = a;
  atomicAdd(&denom[d], a);
}

// =========================================================================
// out[dst] += (a/denom[dst]) * hl[src]  (one wave per edge, float4 per lane)
// =========================================================================
__global__ __launch_bounds__(256)
void gat_edge_scatter(const float* __restrict__ hl, const float* __restrict__ ebuf,
                      const float* __restrict__ denom, const int* __restrict__ ei,
                      float* __restrict__ out, int E, int Etot) {
  const int eid = blockIdx.x * 8 + (threadIdx.x >> 5);
  if (eid >= Etot) return;
  const int lane = threadIdx.x & 31;
  int s, d;
  if (eid < E) { s = ei[eid]; d = ei[E + eid]; }
  else         { s = eid - E; d = s; }

  const float alpha = ebuf[eid] / denom[d];
  const float4 l4 = ((const float4*)(hl + (size_t)s * F))[lane];
  float* o = out + (size_t)d * F + lane * 4;
  atomicAdd(o + 0, alpha * l4.x);
  atomicAdd(o + 1, alpha * l4.y);
  atomicAdd(o + 2, alpha * l4.z);
  atomicAdd(o + 3, alpha * l4.w);
}

// =========================================================================
// h += bias (broadcast over rows), optional ReLU.  float4 per thread.
// =========================================================================
__global__ void gat_bias_act(float* __restrict__ h, const float* __restrict__ b,
                             int n4, int relu) {
  const int i = blockIdx.x * blockDim.x + threadIdx.x;
  if (i >= n4) return;
  float4 v = ((float4*)h)[i];
  const float4 bb = ((const float4*)b)[i & 31];  // 128/4 = 32 float4 of bias
  v.x += bb.x; v.y += bb.y; v.z += bb.z; v.w += bb.w;
  if (relu) {
    v.x = fmaxf(v.x, 0.f); v.y = fmaxf(v.y, 0.f);
    v.z = fmaxf(v.z, 0.f); v.w = fmaxf(v.w, 0.f);
  }
  ((float4*)h)[i] = v;
}

// =========================================================================
// Host-side: one GATv2 layer
// =========================================================================
static void gat_layer(const float* hin, const float* Wl, const float* Wr,
                      const float* att, const float* b, const int* ei,
                      float* hl, float* hr, float* ebuf, unsigned* m, float* denom,
                      float* out, int N, int E, int Etot, int relu,
                      hipStream_t stream) {
  const int tiles = (N + 15) / 16;
  const dim3 tgrid((tiles + 7) / 8);
  gat_wmma_transform<<<tgrid, 256, 0, stream>>>(hin, Wl, hl, N);
  gat_wmma_transform<<<tgrid, 256, 0, stream>>>(hin, Wr, hr, N);

  const int n4 = N * (F / 4);
  gat_zero_f4<<<(n4 + 255) / 256, 256, 0, stream>>>((float4*)out, n4);
  gat_init_nodes<<<(N + 255) / 256, 256, 0, stream>>>(m, denom, N);

  gat_edge_logits<<<(Etot + 7) / 8, 256, 0, stream>>>(hl, hr, ei, att, ebuf, m, E, Etot);
  gat_edge_expsum<<<(Etot + 255) / 256, 256, 0, stream>>>(ebuf, m, ei, denom, E, Etot);
  gat_edge_scatter<<<(Etot + 7) / 8, 256, 0, stream>>>(hl, ebuf, denom, ei, out, E, Etot);

  gat_bias_act<<<(n4 + 255) / 256, 256, 0, stream>>>(out, b, n4, relu);
}

extern "C" void kernel_launch(void* const* d_in, const int* in_sizes, int n_in,
                              void* d_out, int out_size, void* d_ws, size_t ws_size,
                              hipStream_t stream) {
  (void)n_in; (void)out_size; (void)ws_size;

  const float* x    = (const float*)d_in[0];
  const float* pos  = (const float*)d_in[1];   // [N,1,128] == [N,128] flat
  const int*   ei   = (const int*)d_in[2];     // row 0: src, row 1: dst
  const float* W1l  = (const float*)d_in[3];
  const float* W1r  = (const float*)d_in[4];
  const float* att1 = (const float*)d_in[5];
  const float* b1   = (const float*)d_in[6];
  const float* W2l  = (const float*)d_in[7];
  const float* W2r  = (const float*)d_in[8];
  const float* att2 = (const float*)d_in[9];
  const float* b2   = (const float*)d_in[10];

  const int N    = in_sizes[0] / F;   // 100000
  const int E    = in_sizes[2] / 2;   // 1600000
  const int Etot = E + N;             // + self loops
  const size_t NF = (size_t)N * F;

  // workspace layout
  float*    hl    = (float*)d_ws;
  float*    hr    = hl + NF;
  float*    hmid  = hr + NF;
  float*    ebuf  = hmid + NF;
  unsigned* m     = (unsigned*)(ebuf + Etot);
  float*    denom = (float*)(m + N);

  float* outx = (float*)d_out;
  float* outp = outx + NF;

  // ---- GAT on x ----
  gat_layer(x,    W1l, W1r, att1, b1, ei, hl, hr, ebuf, m, denom, hmid, N, E, Etot, 1, stream);
  gat_layer(hmid, W2l, W2r, att2, b2, ei, hl, hr, ebuf, m, denom, outx, N, E, Etot, 0, stream);
  // ---- GAT on pos embeddings ----
  gat_layer(pos,  W1l, W1r, att1, b1, ei, hl, hr, ebuf, m, denom, hmid, N, E, Etot, 1, stream);
  gat_layer(hmid, W2l, W2r, att2, b2, ei, hl, hr, ebuf, m, denom, outp, N, E, Etot, 0, stream);
}